// GAL_Nhop_68032281968811
// MI455X (gfx1250) — compile-verified
//
#include <hip/hip_runtime.h>

typedef float v2f __attribute__((ext_vector_type(2)));
typedef float v8f __attribute__((ext_vector_type(8)));

#define H 128

// ---------------- degree / normalization ----------------

__global__ __launch_bounds__(256) void k_init_deg(float* __restrict__ deg, int n) {
  int i = blockIdx.x * 256 + threadIdx.x;
  if (i < n) deg[i] = 1.0f;  // self loop
}

__global__ __launch_bounds__(256) void k_count_deg(const long long* __restrict__ dst,
                                                   float* __restrict__ deg, int E) {
  int e = blockIdx.x * 256 + threadIdx.x;
  if (e < E) atomicAdd(&deg[dst[e]], 1.0f);
}

__global__ __launch_bounds__(256) void k_dis(float* __restrict__ deg, int n) {
  int i = blockIdx.x * 256 + threadIdx.x;
  if (i < n) deg[i] = rsqrtf(deg[i]);  // deg >= 1 always (self loop)
}

__global__ __launch_bounds__(256) void k_edge_norm(const long long* __restrict__ src,
                                                   const long long* __restrict__ dst,
                                                   const float* __restrict__ dis,
                                                   float* __restrict__ nrm, int E) {
  int e = blockIdx.x * 256 + threadIdx.x;
  if (e < E) nrm[e] = dis[src[e]] * dis[dst[e]];
}

// ---------------- fp32 WMMA GEMM: Y = X @ W^T ----------------
// One wave computes a 16x128 output stripe: 8 column tiles x 32 K-chunks of
// V_WMMA_F32_16X16X4_F32. A (16x4 f32): lanes 0-15 hold M=0..15 {K0,K1},
// lanes 16-31 hold M=0..15 {K2,K3}. B (4x16) mirrors with N in place of M,
// and B[k][n] = W[n][k] gives X@W^T directly. D: VGPR v = row v (+8 for the
// upper lane half), N = lane%16.

__global__ __launch_bounds__(256) void k_gemm_xwT(const float* __restrict__ X,
                                                  const float* __restrict__ W,
                                                  float* __restrict__ Y, int nrows) {
  int wave = blockIdx.x * 8 + (threadIdx.x >> 5);
  int lane = threadIdx.x & 31;
  int row0 = wave << 4;
  if (row0 >= nrows) return;           // wave-uniform: EXEC stays all-1s in body

  int r  = lane & 15;                  // M (for A) / N (for B) within tile
  int kh = (lane >> 4) << 1;           // K sub-offset: 0 or 2

  v8f acc[8];
  const v8f vzero = {0.f, 0.f, 0.f, 0.f, 0.f, 0.f, 0.f, 0.f};
#pragma unroll
  for (int c = 0; c < 8; ++c) acc[c] = vzero;

  const float* xrow = X + (size_t)(row0 + r) * H;
#pragma unroll 4
  for (int kk = 0; kk < H; kk += 4) {
    v2f a = *(const v2f*)(xrow + kk + kh);
#pragma unroll
    for (int c = 0; c < 8; ++c) {
      v2f b = *(const v2f*)(W + (size_t)(c * 16 + r) * H + kk + kh);
      acc[c] = __builtin_amdgcn_wmma_f32_16x16x4_f32(
          /*neg_a=*/false, a, /*neg_b=*/false, b,
          /*c_mod=*/(short)0, acc[c], /*reuse_a=*/false, /*reuse_b=*/false);
    }
  }

  int mbase = (lane >> 4) << 3;        // 0 or 8
#pragma unroll
  for (int c = 0; c < 8; ++c) {
    int col = c * 16 + r;
#pragma unroll
    for (int v = 0; v < 8; ++v) {
      Y[(size_t)(row0 + mbase + v) * H + col] = acc[c][v];
    }
  }
}

// ---------------- self-loop term + bias ----------------

__global__ __launch_bounds__(256) void k_self(const float* __restrict__ X,
                                              const float* __restrict__ dis,
                                              const float* __restrict__ b,
                                              float* __restrict__ out, int n) {
  int t = blockIdx.x * 256 + threadIdx.x;
  if (t >= n * H) return;
  int i = t >> 7;
  int f = t & (H - 1);
  float d = dis[i];
  out[t] = X[t] * d * d + b[f];
}

// ---------------- edge scatter-add (coalesced over 128 features) ----------------

__global__ __launch_bounds__(256) void k_scatter(const long long* __restrict__ src,
                                                 const long long* __restrict__ dst,
                                                 const float* __restrict__ nrm,
                                                 const float* __restrict__ X,
                                                 float* __restrict__ out, int E) {
  int t = blockIdx.x * 256 + threadIdx.x;   // E*H = 76.8M fits in int
  int e = t >> 7;
  if (e >= E) return;
  int f = t & (H - 1);
  long long s = src[e];
  long long d = dst[e];
  float w = nrm[e];
  atomicAdd(&out[d * (long long)H + f], w * X[s * (long long)H + f]);
}

// ---------------- launcher ----------------

extern "C" void kernel_launch(void* const* d_in, const int* in_sizes, int n_in,
                              void* d_out, int out_size, void* d_ws, size_t ws_size,
                              hipStream_t stream) {
  (void)n_in; (void)out_size; (void)ws_size;

  const float*     emb = (const float*)d_in[0];
  const long long* ei  = (const long long*)d_in[1];
  const float*     W1  = (const float*)d_in[2];
  const float*     b1  = (const float*)d_in[3];
  const float*     W2  = (const float*)d_in[4];
  const float*     b2  = (const float*)d_in[5];
  float*           out = (float*)d_out;

  int n = in_sizes[0] / H;     // 50000
  int E = in_sizes[1] / 2;     // 600000
  const long long* srcp = ei;
  const long long* dstp = ei + E;

  char* ws = (char*)d_ws;
  float* dis = (float*)ws;  ws += (((size_t)n * 4) + 255) & ~(size_t)255;
  float* nrm = (float*)ws;  ws += (((size_t)E * 4) + 255) & ~(size_t)255;
  float* X   = (float*)ws;  // n*H floats (reused for both layers' GEMM output)

  int gN  = (n + 255) / 256;
  int gE  = (E + 255) / 256;
  int gNH = (n * H + 255) / 256;
  int gEH = (int)(((long long)E * H + 255) / 256);
  int gG  = ((n + 15) / 16 + 7) / 8;   // 8 waves/block, one 16-row stripe each

  // normalization (shared by both layers)
  k_init_deg <<<gN, 256, 0, stream>>>(dis, n);
  k_count_deg<<<gE, 256, 0, stream>>>(dstp, dis, E);
  k_dis      <<<gN, 256, 0, stream>>>(dis, n);
  k_edge_norm<<<gE, 256, 0, stream>>>(srcp, dstp, dis, nrm, E);

  // layer 1: X = emb @ W1^T ; out = dis^2*X + b1 ; out += scatter(nrm * X[src])
  k_gemm_xwT <<<gG,  256, 0, stream>>>(emb, W1, X, n);
  k_self     <<<gNH, 256, 0, stream>>>(X, dis, b1, out, n);
  k_scatter  <<<gEH, 256, 0, stream>>>(srcp, dstp, nrm, X, out, E);

  // layer 2: X = out @ W2^T ; out = dis^2*X + b2 ; out += scatter(nrm * X[src])
  k_gemm_xwT <<<gG,  256, 0, stream>>>(out, W2, X, n);
  k_self     <<<gNH, 256, 0, stream>>>(X, dis, b2, out, n);
  k_scatter  <<<gEH, 256, 0, stream>>>(srcp, dstp, nrm, X, out, E);
}